// SCAttDec_50586124812564
// MI455X (gfx1250) — compile-verified
//
#include <hip/hip_runtime.h>
#include <math.h>

// Single-query attention decode, b=64 h=8 k=2048 d=128, all f32.
// One workgroup per (b,h); 256 threads = 8 wave32 waves.
// Phase 1: scores via chained V_WMMA_F32_16X16X4_F32 (16 k-rows per wave-tile).
// Phase 2: bandwidth-optimal coalesced VALU streaming of value2 + prefetch.

typedef __attribute__((ext_vector_type(2))) float v2f;
typedef __attribute__((ext_vector_type(8))) float v8f;

#define SC_NEG_INF (-1000000000.0f)

namespace {
constexpr int Bsz = 64;
constexpr int Hsz = 8;
constexpr int Ksz = 2048;
constexpr int Dsz = 128;
constexpr int NT  = 256;   // 8 waves
}

__global__ __launch_bounds__(NT) void scatt_dec_kernel(
    const float* __restrict__ query,     // (B,H,D)
    const float* __restrict__ key,       // (B,H,K,D)
    const int*   __restrict__ att_mask,  // (B,K)
    const float* __restrict__ value1,    // (B,H,D)
    const float* __restrict__ value2,    // (B,H,K,D)
    const float* __restrict__ p_weights, // (1,H,1,D)
    const float* __restrict__ p_bias,    // (1,H,1,1)
    float*       __restrict__ out)       // (B,H,D)
{
  __shared__ float qw[Dsz];      // pw * q
  __shared__ float sc[Ksz];      // scores -> exp weights
  __shared__ float red[NT];      // block reductions
  __shared__ float part[NT];     // phase-2 partials

  const int t    = threadIdx.x;
  const int lane = t & 31;
  const int wave = t >> 5;
  const int bh   = blockIdx.x;   // 0..511
  const int b    = bh >> 3;
  const int h    = bh & 7;

  // ---- qw[d] = p_weights[h,d] * query[b,h,d] ----
  const float* qrow  = query     + (size_t)bh * Dsz;
  const float* pwrow = p_weights + (size_t)h  * Dsz;
  if (t < Dsz) qw[t] = pwrow[t] * qrow[t];
  __syncthreads();

  // ---- Phase 1: scores[k] = dot(qw, key[b,h,k,:]) via WMMA f32 16x16x4 ----
  // D = A x B, A(16x4): every row = qw[dc..dc+3]  -> D[m,n] independent of m
  // B(4x16):  B[kk,n] = key[k0+n, dc+kk]          -> D[*,n] = score partial of row k0+n
  const float* kbase = key + (size_t)bh * Ksz * Dsz;
  const int n    = lane & 15;
  const int koff = (lane >> 4) * 2;   // lanes 0-15: K{0,1}; lanes 16-31: K{2,3}
#if __has_builtin(__builtin_amdgcn_wmma_f32_16x16x4_f32)
  for (int tile = wave; tile < Ksz / 16; tile += 8) {
    const int k0 = tile * 16;
    const float* krow = kbase + (size_t)(k0 + n) * Dsz;
    v8f c = {0.f, 0.f, 0.f, 0.f, 0.f, 0.f, 0.f, 0.f};
    #pragma unroll 8
    for (int dc = 0; dc < Dsz; dc += 4) {
      v2f a, bm;
      a.x  = qw[dc + koff];
      a.y  = qw[dc + koff + 1];
      bm.x = krow[dc + koff];
      bm.y = krow[dc + koff + 1];
      c = __builtin_amdgcn_wmma_f32_16x16x4_f32(
              /*neg_a=*/false, a, /*neg_b=*/false, bm,
              /*c_mod=*/(short)0, c, /*reuse_a=*/false, /*reuse_b=*/false);
    }
    if (lane < 16) sc[k0 + n] = c[0];   // lane n holds score[k0+n] in every D elem
  }
#else
  for (int k = t; k < Ksz; k += NT) {   // fallback (histogram would show wmma=0)
    const float* krow = kbase + (size_t)k * Dsz;
    float s = 0.f;
    for (int d = 0; d < Dsz; ++d) s += qw[d] * krow[d];
    sc[k] = s;
  }
#endif
  __syncthreads();

  // ---- mask + bias ----
  const float pb   = p_bias[h];
  const int*  mrow = att_mask + (size_t)b * Ksz;
  for (int k = t; k < Ksz; k += NT) {
    const float s = sc[k] + pb;
    sc[k] = (mrow[k] == 0) ? SC_NEG_INF : s;
  }
  __syncthreads();

  // ---- softmax: block max ----
  float m = SC_NEG_INF;
  for (int k = t; k < Ksz; k += NT) m = fmaxf(m, sc[k]);
  red[t] = m;
  __syncthreads();
  for (int off = NT / 2; off > 0; off >>= 1) {
    if (t < off) red[t] = fmaxf(red[t], red[t + off]);
    __syncthreads();
  }
  const float rowmax = red[0];
  __syncthreads();

  // ---- softmax: exp + block sum (sc[] becomes unnormalized alpha) ----
  float lsum = 0.f;
  for (int k = t; k < Ksz; k += NT) {
    const float e = __expf(sc[k] - rowmax);
    sc[k] = e;
    lsum += e;
  }
  red[t] = lsum;
  __syncthreads();
  for (int off = NT / 2; off > 0; off >>= 1) {
    if (t < off) red[t] += red[t + off];
    __syncthreads();
  }
  const float inv = 1.f / red[0];
  __syncthreads();

  // ---- Phase 2: acc[d] = sum_k alpha[k] * value2[b,h,k,d] ----
  // threads 0-127 -> d=t, even k; threads 128-255 -> d=t-128, odd k.
  // 128 consecutive threads read one contiguous 512B row segment: fully coalesced.
  const float* v2base = value2 + (size_t)bh * Ksz * Dsz;
  const int d    = t & (Dsz - 1);
  const int half = t >> 7;
  float a0 = 0.f, a1 = 0.f, a2 = 0.f, a3 = 0.f;
  for (int k = half; k < Ksz; k += 8) {
    const float* p0 = v2base + (size_t)k * Dsz + d;
    const float w0 = sc[k], w1 = sc[k + 2], w2 = sc[k + 4], w3 = sc[k + 6];
    a0 = fmaf(w0, p0[0 * Dsz], a0);
    a1 = fmaf(w1, p0[2 * Dsz], a1);
    a2 = fmaf(w2, p0[4 * Dsz], a2);
    a3 = fmaf(w3, p0[6 * Dsz], a3);
    __builtin_prefetch(p0 + 16 * Dsz, 0, 0);   // global_prefetch_b8, 16 rows ahead
  }
  part[t] = (a0 + a1) + (a2 + a3);
  __syncthreads();

  if (t < Dsz) {
    const float tot = (part[t] + part[t + 128]) * inv;
    out[(size_t)bh * Dsz + t] = value1[(size_t)bh * Dsz + t] * tot;
  }
}

extern "C" void kernel_launch(void* const* d_in, const int* in_sizes, int n_in,
                              void* d_out, int out_size, void* d_ws, size_t ws_size,
                              hipStream_t stream) {
  (void)in_sizes; (void)n_in; (void)out_size; (void)d_ws; (void)ws_size;
  const float* query     = (const float*)d_in[0];
  const float* key       = (const float*)d_in[1];
  const int*   att_mask  = (const int*)  d_in[2];
  const float* value1    = (const float*)d_in[3];
  const float* value2    = (const float*)d_in[4];
  const float* p_weights = (const float*)d_in[5];
  const float* p_bias    = (const float*)d_in[6];
  float* out = (float*)d_out;

  dim3 grid(Bsz * Hsz);   // 512 blocks, one per (b,h)
  dim3 block(NT);         // 8 wave32 waves
  scatt_dec_kernel<<<grid, block, 0, stream>>>(
      query, key, att_mask, value1, value2, p_weights, p_bias, out);
}